// _GroundingDinoMultiscaleDeformableAttention_53343493816658
// MI455X (gfx1250) — compile-verified
//
#include <hip/hip_runtime.h>
#include <math.h>

typedef float v2f __attribute__((ext_vector_type(2)));
typedef float v8f __attribute__((ext_vector_type(8)));

#define D_MODEL 256
#define N_HEADS 8
#define D_HEAD 32
#define N_LEVELS 4
#define N_POINTS 4
#define GEMM_K 256            // all four GEMMs have K = D_MODEL = 256
#define MT_PER_BLOCK 8        // 8 waves per block, one m-tile each

// ---------------------------------------------------------------------------
// hs = a + b   (vectorized float4)
// ---------------------------------------------------------------------------
__global__ void add_kernel(const float* __restrict__ a, const float* __restrict__ b,
                           float* __restrict__ c, int n4) {
  int i = blockIdx.x * blockDim.x + threadIdx.x;
  if (i < n4) {
    float4 x = ((const float4*)a)[i];
    float4 y = ((const float4*)b)[i];
    ((float4*)c)[i] = make_float4(x.x + y.x, x.y + y.y, x.z + y.z, x.w + y.w);
  }
}

// ---------------------------------------------------------------------------
// C[M,N] = A[M,K=256] * W[256,N] + bias[N]   via V_WMMA_F32_16X16X4_F32
//
// Block = 256 threads = 8 waves = 8 consecutive m-tiles x 1 n-tile (16 cols).
// The 256x16 W strip is staged once into LDS in WMMA-operand order:
//   Bsh[(kpair*16 + col)*2 + (k&1)] = W[k][n0+col]
// so each wave's B operand is a single conflict-free ds_load_b64:
//   lane l (pair = kk/2 + l/16, col = l%16) reads rows {2*pair, 2*pair+1}.
//
// f32 WMMA register layouts (ISA 7.12.2):
//   A 16x4: lane = M%16, lane-half selects K base {0,2}, reg +v.
//   B 4x16: lane = N%16, lane-half selects K base {0,2}, reg +v.
//   C/D:    reg r -> row m0 + r + (lane>=16 ? 8 : 0), col n0 + lane%16.
// ---------------------------------------------------------------------------
__global__ void gemm_wmma_f32(const float* __restrict__ A, const float* __restrict__ W,
                              const float* __restrict__ bias, float* __restrict__ C,
                              int M, int N) {
  __shared__ float Bsh[GEMM_K * 16];   // 16 KB

  int ntiles  = N >> 4;
  int nt      = blockIdx.x % ntiles;
  int mb      = blockIdx.x / ntiles;
  int n0      = nt << 4;

  // ---- cooperative stage of W[:, n0:n0+16] into LDS (interleaved k-pairs) ----
  {
    int t  = threadIdx.x;
    int c  = t & 15;       // column 0..15
    int k0 = t >> 4;       // 0..15
    #pragma unroll
    for (int j = 0; j < 16; ++j) {
      int k = k0 + (j << 4);                       // 0..255
      Bsh[(((k >> 1) << 4) + c) * 2 + (k & 1)] = W[(size_t)k * N + n0 + c];
    }
  }
  __syncthreads();

  // ---- per-wave 16x16 tile ----
  int lane   = threadIdx.x & 31;
  int wv     = threadIdx.x >> 5;                   // 0..7
  int mtiles = (M + 15) >> 4;
  int mt     = mb * MT_PER_BLOCK + wv;
  if (mt >= mtiles) return;                        // wave-uniform (EXEC stays full)

  int m0 = mt << 4;
  int ra = m0 + (lane & 15);
  if (ra > M - 1) ra = M - 1;                      // clamp tail rows; stores predicated
  int kh = (lane >> 4) << 1;                       // K base 0 or 2

  const float* Ap  = A + (size_t)ra * GEMM_K + kh;
  const float* Bp  = &Bsh[(((kh >> 1) << 4) + (lane & 15)) * 2]; // pair offset for lane-half

  v8f acc = {};
  #pragma unroll 8
  for (int kk = 0; kk < GEMM_K; kk += 4) {
    v2f a = *(const v2f*)(Ap + kk);                // global_load_b64
    v2f b = *(const v2f*)(Bp + (kk << 4));         // ds_load_b64 (kk/2 pairs * 32 floats)
    acc = __builtin_amdgcn_wmma_f32_16x16x4_f32(
        /*neg_a=*/false, a, /*neg_b=*/false, b,
        /*c_mod=*/(short)0, acc, /*reuse_a=*/false, /*reuse_b=*/false);
  }

  float bv  = bias[n0 + (lane & 15)];
  int rbase = m0 + ((lane >> 4) << 3);
  float* Cp = C + (size_t)rbase * N + n0 + (lane & 15);
  #pragma unroll
  for (int r = 0; r < 8; ++r) {
    if (rbase + r < M) Cp[(size_t)r * N] = acc[r] + bv;
  }
}

// ---------------------------------------------------------------------------
// In-place softmax over contiguous groups of 16 (per (row, head)).
// ---------------------------------------------------------------------------
__global__ void softmax16_kernel(float* __restrict__ g, int ngroups) {
  int i = blockIdx.x * blockDim.x + threadIdx.x;
  if (i >= ngroups) return;
  float* p = g + (size_t)i * 16;
  float v[16];
  float m = -3.402823466e+38f;
  #pragma unroll
  for (int j = 0; j < 16; ++j) { v[j] = p[j]; m = fmaxf(m, v[j]); }
  float s = 0.f;
  #pragma unroll
  for (int j = 0; j < 16; ++j) { v[j] = __expf(v[j] - m); s += v[j]; }
  float r = 1.f / s;
  #pragma unroll
  for (int j = 0; j < 16; ++j) p[j] = v[j] * r;
}

// ---------------------------------------------------------------------------
// Deformable bilinear sampling + attention-weighted sum.
// One wave32 per (b, q, head); lane = d_head channel -> every corner fetch is
// a coalesced 128B read (value tensor is L2-resident: 27MB << 192MB L2).
// ---------------------------------------------------------------------------
__global__ void deform_sample_kernel(const float* __restrict__ value,
                                     const float* __restrict__ offs,
                                     const float* __restrict__ attnw,
                                     const float* __restrict__ refp,
                                     const int*   __restrict__ shapes,
                                     float* __restrict__ samp,
                                     int B, int Q) {
  int lane = threadIdx.x & 31;
  int gw   = (int)((blockIdx.x * blockDim.x + threadIdx.x) >> 5);
  int total = B * Q * N_HEADS;
  if (gw >= total) return;

  int h  = gw & (N_HEADS - 1);
  int bq = gw >> 3;
  int b  = bq / Q;

  const float* offp = offs  + (size_t)bq * 256 + h * (N_LEVELS * N_POINTS * 2);
  const float* ap   = attnw + (size_t)bq * 128 + h * (N_LEVELS * N_POINTS);
  const float* rp   = refp  + (size_t)bq * (N_LEVELS * 2);

  float acc = 0.f;
  int start = 0;
  for (int l = 0; l < N_LEVELS; ++l) {
    int   hl = shapes[2 * l];
    int   wl = shapes[2 * l + 1];
    float xb = rp[2 * l]     * (float)wl - 0.5f;   // ref_x*w - 0.5
    float yb = rp[2 * l + 1] * (float)hl - 0.5f;   // ref_y*h - 0.5
    const float* vbase = value + (size_t)(b * Q + start) * 256 + h * D_HEAD + lane;

    auto corner = [&](int yi, int xi, float wgt) -> float {
      if (yi < 0 || yi >= hl || xi < 0 || xi >= wl) return 0.f;  // zero padding
      return vbase[(size_t)(yi * wl + xi) * 256] * wgt;
    };

    #pragma unroll
    for (int p = 0; p < N_POINTS; ++p) {
      float x   = xb + offp[(l * N_POINTS + p) * 2];
      float y   = yb + offp[(l * N_POINTS + p) * 2 + 1];
      float x0f = floorf(x), y0f = floorf(y);
      int   x0  = (int)x0f,  y0  = (int)y0f;
      float lx  = x - x0f,   ly  = y - y0f;
      float aw  = ap[l * N_POINTS + p];

      float s = corner(y0,     x0,     (1.f - ly) * (1.f - lx))
              + corner(y0,     x0 + 1, (1.f - ly) * lx)
              + corner(y0 + 1, x0,     ly * (1.f - lx))
              + corner(y0 + 1, x0 + 1, ly * lx);
      acc += aw * s;
    }
    start += hl * wl;
  }
  samp[(size_t)bq * 256 + h * D_HEAD + lane] = acc;
}

// ---------------------------------------------------------------------------
// Launch
// Inputs: 0 hidden, 1 pos, 2 enc, 3 ref_points, 4 spatial_shapes(int),
//         5 W_value, 6 b_value, 7 W_off, 8 b_off, 9 W_attn, 10 b_attn,
//         11 W_out, 12 b_out
// Output: [ out (M*256) | attn (M*128) ]  floats
// ---------------------------------------------------------------------------
extern "C" void kernel_launch(void* const* d_in, const int* in_sizes, int n_in,
                              void* d_out, int out_size, void* d_ws, size_t ws_size,
                              hipStream_t stream) {
  const float* hidden = (const float*)d_in[0];
  const float* pos    = (const float*)d_in[1];
  const float* enc    = (const float*)d_in[2];
  const float* refp   = (const float*)d_in[3];
  const int*   shapes = (const int*)d_in[4];
  const float* Wv     = (const float*)d_in[5];
  const float* bv     = (const float*)d_in[6];
  const float* Woff   = (const float*)d_in[7];
  const float* boff   = (const float*)d_in[8];
  const float* Wa     = (const float*)d_in[9];
  const float* ba     = (const float*)d_in[10];
  const float* Wout   = (const float*)d_in[11];
  const float* bout   = (const float*)d_in[12];

  const int B = 2;
  const int M = in_sizes[0] / D_MODEL;   // B*Q rows
  const int Q = M / B;

  float* out  = (float*)d_out;                          // M*256
  float* attn = (float*)d_out + (size_t)M * D_MODEL;    // M*128 (second output)

  float* hs   = (float*)d_ws;                           // M*256 (later reused as samp)
  float* val  = hs  + (size_t)M * D_MODEL;              // M*256
  float* offs = val + (size_t)M * D_MODEL;              // M*256

  const int mtiles  = (M + 15) / 16;
  const int mblocks = (mtiles + MT_PER_BLOCK - 1) / MT_PER_BLOCK;

  // 1. hs = hidden + pos
  {
    int n4 = (M * D_MODEL) / 4;
    add_kernel<<<(n4 + 255) / 256, 256, 0, stream>>>(hidden, pos, hs, n4);
  }
  // 2. value = enc @ Wv + bv   (N=256)
  gemm_wmma_f32<<<mblocks * (D_MODEL / 16), 256, 0, stream>>>(enc, Wv, bv, val, M, D_MODEL);
  // 3. offs = hs @ Woff + boff (N=256)
  gemm_wmma_f32<<<mblocks * (D_MODEL / 16), 256, 0, stream>>>(hs, Woff, boff, offs, M, D_MODEL);
  // 4. attn_raw = hs @ Wa + ba (N=128) -> straight into d_out attn region
  gemm_wmma_f32<<<mblocks * (128 / 16), 256, 0, stream>>>(hs, Wa, ba, attn, M, 128);
  // 5. softmax in place (groups of 16 = per head L*P)
  {
    int ng = M * N_HEADS;
    softmax16_kernel<<<(ng + 255) / 256, 256, 0, stream>>>(attn, ng);
  }
  // 6. deformable sampling -> samp (reuse hs buffer)
  {
    int waves = M * N_HEADS;
    deform_sample_kernel<<<(waves + 7) / 8, 256, 0, stream>>>(val, offs, attn, refp,
                                                              shapes, hs, B, Q);
  }
  // 7. out = samp @ Wout + bout (N=256)
  gemm_wmma_f32<<<mblocks * (D_MODEL / 16), 256, 0, stream>>>(hs, Wout, bout, out, M, D_MODEL);
}